// GFC_60979945668772
// MI455X (gfx1250) — compile-verified
//
#include <hip/hip_runtime.h>
#include <math.h>

// CDNA5 / gfx1250: fp32 Gram via V_WMMA_F32_16X16X4_F32, wave32, 8 waves/block.

typedef __attribute__((ext_vector_type(2))) float v2f;
typedef __attribute__((ext_vector_type(8))) float v8f;

#define N_    32
#define C_    64
#define T_    256
#define F_    64
#define TK    128           // K chunk staged in LDS
#define LDSW  (TK + 2)      // +2 floats row padding -> conflict-free ds_load_b64
#define TRIU  2016          // C*(C-1)/2
#define KIDX  1008          // nearest-quantile index: round(0.5*(2016-1))

__global__ __launch_bounds__(256) void gram_rbf_kernel(
    const float* __restrict__ X,        // (N, C, T, F)
    const float* __restrict__ gammad,   // scalar
    float* __restrict__ out)            // (N, F, C, C)
{
    __shared__ float ldsX[C_ * LDSW];   // 33,280 B; reused as sort buffer later
    __shared__ float Dm[C_ * 65];       // 16,640 B; G then D, padded stride 65
    __shared__ float diag[C_];

    const int tid = threadIdx.x;
    const int bid = blockIdx.x;         // n*F + f
    const int n   = bid >> 6;
    const int f   = bid & 63;

    const float* Xg = X + ((size_t)n * C_ * T_ * F_) + f;

    // ---- per-wave WMMA tile assignment -------------------------------------
    const int lane = tid & 31;
    const int wave = tid >> 5;
    const int r    = lane & 15;
    const int koff = (lane >> 4) << 1;          // lanes 0-15: K 0,1 ; 16-31: K 2,3

    const int tile0 = wave * 2;                 // two tiles share the A row block
    const int m0  = (tile0 >> 2) << 4;
    const int n0a = (tile0 & 3) << 4;
    const int n0b = n0a + 16;

    v8f acc0 = {};
    v8f acc1 = {};

    // ---- Gram over T in two K-chunks of 128 --------------------------------
    for (int chunk = 0; chunk < 2; ++chunk) {
        __syncthreads();  // protect LDS reuse across chunks
        // stage Xp[c, t] = X[n, c, chunk*128 + t, f]
        for (int idx = tid; idx < C_ * TK; idx += 256) {
            int c = idx >> 7;
            int t = idx & (TK - 1);
            ldsX[c * LDSW + t] =
                Xg[((size_t)(c * T_ + chunk * TK + t)) * F_];
        }
        __syncthreads();

        const float* rowA  = &ldsX[(m0  + r) * LDSW + koff];
        const float* rowB0 = &ldsX[(n0a + r) * LDSW + koff];
        const float* rowB1 = &ldsX[(n0b + r) * LDSW + koff];

        #pragma unroll 4
        for (int kb = 0; kb < TK; kb += 4) {
            v2f a  = *(const v2f*)(rowA  + kb);
            v2f b0 = *(const v2f*)(rowB0 + kb);
            v2f b1 = *(const v2f*)(rowB1 + kb);
            acc0 = __builtin_amdgcn_wmma_f32_16x16x4_f32(
                       false, a, false, b0, (short)0, acc0, false, false);
            acc1 = __builtin_amdgcn_wmma_f32_16x16x4_f32(
                       false, a, false, b1, (short)0, acc1, false, false);
        }
    }

    // ---- store G tiles to LDS (C/D layout: VGPR v -> row m0+v (+8 hi half)) --
    {
        int rowBase = m0 + ((lane >> 4) << 3);
        int col0 = n0a + r;
        int col1 = n0b + r;
        #pragma unroll
        for (int v = 0; v < 8; ++v) {
            Dm[(rowBase + v) * 65 + col0] = acc0[v];
            Dm[(rowBase + v) * 65 + col1] = acc1[v];
        }
    }
    __syncthreads();

    if (tid < C_) diag[tid] = Dm[tid * 65 + tid];
    __syncthreads();

    // ---- D = diag_i + diag_j - 2G ; gather strict upper triangle -----------
    float* s = ldsX;  // staging LDS no longer needed: reuse as 2048-float sort buf
    for (int p = tid; p < C_ * C_; p += 256) {
        int i = p >> 6;
        int j = p & 63;
        float dv = diag[i] + diag[j] - 2.0f * Dm[i * 65 + j];
        Dm[i * 65 + j] = dv;
        if (i < j) {
            int slot = 63 * i - ((i * (i - 1)) >> 1) + (j - i - 1);
            s[slot] = dv;
        }
    }
    if (tid < 2048 - TRIU) s[TRIU + tid] = __builtin_inff();
    __syncthreads();

    // ---- exact median via bitonic sort of 2048 (ascending) -----------------
    for (int k = 2; k <= 2048; k <<= 1) {
        for (int j = k >> 1; j > 0; j >>= 1) {
            for (int i = tid; i < 2048; i += 256) {
                int ixj = i ^ j;
                if (ixj > i) {
                    float a = s[i];
                    float b = s[ixj];
                    bool up = ((i & k) == 0);
                    if (up ? (a > b) : (a < b)) {
                        s[i]   = b;
                        s[ixj] = a;
                    }
                }
            }
            __syncthreads();
        }
    }

    // sigma^2 == median D value (sqrt is monotone, 'nearest' picks an element)
    float med   = s[KIDX];
    float denom = 2.0f * exp10f(gammad[0]) * med;
    float ninv  = -1.0f / denom;

    float* outp = out + ((size_t)bid << 12);   // * C_*C_
    for (int p = tid; p < C_ * C_; p += 256) {
        outp[p] = expf(Dm[(p >> 6) * 65 + (p & 63)] * ninv);
    }
}

extern "C" void kernel_launch(void* const* d_in, const int* in_sizes, int n_in,
                              void* d_out, int out_size, void* d_ws, size_t ws_size,
                              hipStream_t stream) {
    (void)in_sizes; (void)n_in; (void)out_size; (void)d_ws; (void)ws_size;
    const float* X      = (const float*)d_in[0];
    const float* gammad = (const float*)d_in[1];
    float* out          = (float*)d_out;
    dim3 grid(N_ * F_);    // 2048 blocks: one per (n, f)
    dim3 block(256);       // 8 wave32 waves
    hipLaunchKernelGGL(gram_rbf_kernel, grid, block, 0, stream, X, gammad, out);
}